// TensorProductUniform1dJit_59356448030870
// MI455X (gfx1250) — compile-verified
//
#include <hip/hip_runtime.h>
#include <cstdint>
#include <cstddef>

// ---------------------------------------------------------------------------
// Quaternion tensor product, B x (4*128) f32, memory-bound streaming kernel.
// CDNA5 path: async global->LDS staging (ASYNCcnt) + double buffering + NT
// stores. WMMA is deliberately not used: intensity is 0.58 flop/byte.
// ---------------------------------------------------------------------------

typedef __attribute__((ext_vector_type(4))) float fvec4;
typedef int ivec4_t __attribute__((vector_size(16)));   // v4i32, matches builtin

#define ROW      512            // NSEG * EXTENT floats per batch row
#define SEG      128            // EXTENT
#define TROWS    8              // batch rows per tile
#define TILE_F   (TROWS * ROW)  // 4096 floats (16 KB) per operand tile
#define NTHREADS 256

// ---- CDNA5 async-copy feature detection -----------------------------------
#if defined(__has_builtin)
#  if __has_builtin(__builtin_amdgcn_global_load_async_to_lds_b128)
#    define HAVE_ASYNC_LDS 1
#  endif
#  if __has_builtin(__builtin_amdgcn_s_wait_asynccnt)
#    define HAVE_WAIT_ASYNC 1
#  endif
#endif

typedef __attribute__((address_space(1))) ivec4_t gbl_v4i_t;   // global v4i32
typedef __attribute__((address_space(3))) ivec4_t lds_v4i_t;   // LDS v4i32

// Flat->AS1: value-preserving. Flat LDS->AS3: low 32 bits are the LDS byte
// offset (ISA 10.2 aperture rules), so truncation yields a valid AS3 pointer.
__device__ __forceinline__ gbl_v4i_t* to_gbl_v4i(const void* p) {
    return (gbl_v4i_t*)(uintptr_t)p;
}
__device__ __forceinline__ lds_v4i_t* to_lds_v4i(const void* p) {
    return (lds_v4i_t*)(uint32_t)(uintptr_t)p;
}

template <int N>
__device__ __forceinline__ void wait_async() {
#if defined(HAVE_WAIT_ASYNC)
    __builtin_amdgcn_s_wait_asynccnt(N);
#elif defined(HAVE_ASYNC_LDS)
    asm volatile("s_wait_asynccnt %0" ::"n"(N) : "memory");
#endif
}

// Copy 16 bytes global -> LDS, per lane.
__device__ __forceinline__ void stage16(const float* __restrict__ gsrc,
                                        float* ldst) {
#if defined(HAVE_ASYNC_LDS)
    __builtin_amdgcn_global_load_async_to_lds_b128(to_gbl_v4i(gsrc),
                                                   to_lds_v4i(ldst),
                                                   /*offset=*/0, /*cpol=*/0);
#else
    *(fvec4*)ldst = __builtin_nontemporal_load((const fvec4*)gsrc);
#endif
}

// Hamilton product, elementwise over a float4 of extent lanes.
__device__ __forceinline__ void quat4(const fvec4& a0, const fvec4& a1,
                                      const fvec4& a2, const fvec4& a3,
                                      const fvec4& b0, const fvec4& b1,
                                      const fvec4& b2, const fvec4& b3,
                                      fvec4& o0, fvec4& o1, fvec4& o2, fvec4& o3) {
    o0 = a0 * b0 - a1 * b1 - a2 * b2 - a3 * b3;
    o1 = a0 * b1 + a1 * b0 + a2 * b3 - a3 * b2;
    o2 = a0 * b2 - a1 * b3 + a2 * b0 + a3 * b1;
    o3 = a0 * b3 + a1 * b2 - a2 * b1 + a3 * b0;
}

__global__ __launch_bounds__(NTHREADS)
void quat_tp_kernel(const float* __restrict__ in0,
                    const float* __restrict__ in1,
                    float* __restrict__ out,
                    int ntiles) {
    // [2 buffers][2 operands][TILE_F floats] = 64 KB
    __shared__ __align__(16) float smem[2 * 2 * TILE_F];

    const int t = threadIdx.x;

    auto stage_tile = [&](int buf, long long tile) {
        const long long base = tile * (long long)TILE_F;
        float* s0 = &smem[(2 * buf + 0) * TILE_F];
        float* s1 = &smem[(2 * buf + 1) * TILE_F];
#pragma unroll
        for (int k = 0; k < 4; ++k) {               // 4 x 4 KB per operand
            const int c = k * NTHREADS + t;         // 16B chunk id, 0..1023
            stage16(in0 + base + (long long)c * 4, s0 + c * 4);
        }
#pragma unroll
        for (int k = 0; k < 4; ++k) {
            const int c = k * NTHREADS + t;
            stage16(in1 + base + (long long)c * 4, s1 + c * 4);
        }
    };

    long long tile = blockIdx.x;
    const long long stride = gridDim.x;
    if (tile >= ntiles) return;

    stage_tile(0, tile);                            // prologue: fill buffer 0
    int buf = 0;

    for (; tile < ntiles; tile += stride) {
        const long long next = tile + stride;
        const bool has_next = next < ntiles;

        if (has_next) {
            stage_tile(buf ^ 1, next);              // prefetch next tile
            wait_async<8>();                        // drain current tile's 8 copies
        } else {
            wait_async<0>();
        }
        __syncthreads();                            // all waves' data resident

        // ---- compute: thread owns float4 e-group g of row r ----
        const int r = t >> 5;                       // 0..7
        const int g = t & 31;                       // 0..31
        const float* s0 = &smem[(2 * buf + 0) * TILE_F + r * ROW + g * 4];
        const float* s1 = &smem[(2 * buf + 1) * TILE_F + r * ROW + g * 4];

        fvec4 a0 = *(const fvec4*)(s0 + 0 * SEG);
        fvec4 a1 = *(const fvec4*)(s0 + 1 * SEG);
        fvec4 a2 = *(const fvec4*)(s0 + 2 * SEG);
        fvec4 a3 = *(const fvec4*)(s0 + 3 * SEG);
        fvec4 b0 = *(const fvec4*)(s1 + 0 * SEG);
        fvec4 b1 = *(const fvec4*)(s1 + 1 * SEG);
        fvec4 b2 = *(const fvec4*)(s1 + 2 * SEG);
        fvec4 b3 = *(const fvec4*)(s1 + 3 * SEG);

        fvec4 o0, o1, o2, o3;
        quat4(a0, a1, a2, a3, b0, b1, b2, b3, o0, o1, o2, o3);

        float* op = out + tile * (long long)TILE_F + r * ROW + g * 4;
        __builtin_nontemporal_store(o0, (fvec4*)(op + 0 * SEG));
        __builtin_nontemporal_store(o1, (fvec4*)(op + 1 * SEG));
        __builtin_nontemporal_store(o2, (fvec4*)(op + 2 * SEG));
        __builtin_nontemporal_store(o3, (fvec4*)(op + 3 * SEG));

        __syncthreads();                            // reads done before next
        buf ^= 1;                                   // iteration restages buf
    }
}

// Tail rows (B % TROWS) handled directly from global memory (B=262144 has
// no tail; kept for generality).
__global__ __launch_bounds__(NTHREADS)
void quat_tp_tail(const float* __restrict__ in0,
                  const float* __restrict__ in1,
                  float* __restrict__ out,
                  long long startRow, long long nRows) {
    const long long idx =
        (long long)blockIdx.x * blockDim.x + threadIdx.x;  // one per e-group
    const long long total = nRows * (SEG / 4);
    if (idx >= total) return;
    const long long row = startRow + idx / (SEG / 4);
    const int g = (int)(idx % (SEG / 4));
    const float* p0 = in0 + row * ROW + g * 4;
    const float* p1 = in1 + row * ROW + g * 4;

    fvec4 a0 = __builtin_nontemporal_load((const fvec4*)(p0 + 0 * SEG));
    fvec4 a1 = __builtin_nontemporal_load((const fvec4*)(p0 + 1 * SEG));
    fvec4 a2 = __builtin_nontemporal_load((const fvec4*)(p0 + 2 * SEG));
    fvec4 a3 = __builtin_nontemporal_load((const fvec4*)(p0 + 3 * SEG));
    fvec4 b0 = __builtin_nontemporal_load((const fvec4*)(p1 + 0 * SEG));
    fvec4 b1 = __builtin_nontemporal_load((const fvec4*)(p1 + 1 * SEG));
    fvec4 b2 = __builtin_nontemporal_load((const fvec4*)(p1 + 2 * SEG));
    fvec4 b3 = __builtin_nontemporal_load((const fvec4*)(p1 + 3 * SEG));

    fvec4 o0, o1, o2, o3;
    quat4(a0, a1, a2, a3, b0, b1, b2, b3, o0, o1, o2, o3);

    float* op = out + row * ROW + g * 4;
    __builtin_nontemporal_store(o0, (fvec4*)(op + 0 * SEG));
    __builtin_nontemporal_store(o1, (fvec4*)(op + 1 * SEG));
    __builtin_nontemporal_store(o2, (fvec4*)(op + 2 * SEG));
    __builtin_nontemporal_store(o3, (fvec4*)(op + 3 * SEG));
}

extern "C" void kernel_launch(void* const* d_in, const int* in_sizes, int n_in,
                              void* d_out, int out_size, void* d_ws,
                              size_t ws_size, hipStream_t stream) {
    (void)n_in; (void)out_size; (void)d_ws; (void)ws_size;
    const float* in0 = (const float*)d_in[0];
    const float* in1 = (const float*)d_in[1];
    float* out = (float*)d_out;

    const long long total = in_sizes[0];        // B * 512
    const long long B = total / ROW;
    const long long ntiles = B / TROWS;
    const long long tailRows = B - ntiles * TROWS;

    if (ntiles > 0) {
        // Enough blocks to fill every WGP several times over, while giving
        // each block >=2 tiles so double buffering engages (32768 tiles total
        // at the reference size -> 4 tiles/block).
        long long blocks = ntiles < 8192 ? ntiles : 8192;
        quat_tp_kernel<<<(int)blocks, NTHREADS, 0, stream>>>(in0, in1, out,
                                                             (int)ntiles);
    }
    if (tailRows > 0) {
        const long long groups = tailRows * (SEG / 4);
        const int tb = NTHREADS;
        const int gb = (int)((groups + tb - 1) / tb);
        quat_tp_tail<<<gb, tb, 0, stream>>>(in0, in1, out, ntiles * TROWS,
                                            tailRows);
    }
}